// i_k1_calc_impl2_50405736186086
// MI455X (gfx1250) — compile-verified
//
#include <hip/hip_runtime.h>
#include <hip/hip_bf16.h>
#include <stdint.h>

#define TILE  256   // rows per tile == blockDim.x
#define DEPTH 4     // async pipeline depth (LDS buffers)
#define TPB   8     // tiles per block

// ---- compile-time constants from the reference ----
#define C_RS   0.018f
#define C_DT   5.0e-5f
#define C_WEL  942.47779607693793f   /* W_EL = 2*3000*pi/60*3 = 300*pi */
#define C_S1   186.66666666666666f   /* U_DC/2 * S_ABC = 280*(2/3); S_BC term is exactly 0 */

// Load a wave-uniform scalar and pin it in an SGPR (cuts VGPR pressure).
__device__ __forceinline__ float uload(const float* p) {
  return __uint_as_float(__builtin_amdgcn_readfirstlane(__float_as_uint(*p)));
}

__device__ __forceinline__ float fsqrt_fast(float x) {
#if __has_builtin(__builtin_amdgcn_sqrtf)
  return __builtin_amdgcn_sqrtf(x);
#else
  return sqrtf(x);
#endif
}

__device__ __forceinline__ float frcp_nr(float x) {
#if __has_builtin(__builtin_amdgcn_rcpf)
  float r = __builtin_amdgcn_rcpf(x);
  return r * __builtin_fmaf(-x, r, 2.0f);   // one Newton step -> ~1 ulp
#else
  return 1.0f / x;
#endif
}

__device__ __forceinline__ float ftanh(float x) {
#if __has_builtin(__builtin_amdgcn_tanhf)
  return __builtin_amdgcn_tanhf(x);          // gfx1250 v_tanh_f32
#elif __has_builtin(__builtin_amdgcn_tanh_f32)
  return __builtin_amdgcn_tanh_f32(x);
#else
  float ax = __builtin_fabsf(x);
#if __has_builtin(__builtin_amdgcn_exp2f)
  float e = __builtin_amdgcn_exp2f(-2.8853900817779268f * ax); // exp(-2|x|)
#else
  float e = exp2f(-2.8853900817779268f * ax);
#endif
  float t = (1.0f - e) * frcp_nr(1.0f + e);
  return __builtin_copysignf(t, x);
#endif
}

struct Wgt {
  float wph[12], bph[4], wp[8], bp[2], wl[12], bl[4];
};

__device__ __forceinline__ void load_weights(
    Wgt& w,
    const float* __restrict__ Wph, const float* __restrict__ bph,
    const float* __restrict__ Wp,  const float* __restrict__ bp,
    const float* __restrict__ WL,  const float* __restrict__ bL) {
#pragma unroll
  for (int k = 0; k < 12; ++k) w.wph[k] = uload(Wph + k);
#pragma unroll
  for (int k = 0; k < 4;  ++k) w.bph[k] = uload(bph + k);
#pragma unroll
  for (int k = 0; k < 8;  ++k) w.wp[k]  = uload(Wp + k);
#pragma unroll
  for (int k = 0; k < 2;  ++k) w.bp[k]  = uload(bp + k);
#pragma unroll
  for (int k = 0; k < 12; ++k) w.wl[k]  = uload(WL + k);
#pragma unroll
  for (int k = 0; k < 4;  ++k) w.bl[k]  = uload(bL + k);
}

// The per-row math, exactly mirroring the reference (with constant folding).
__device__ __forceinline__ float2 compute_row(float id, float iq, float u2, const Wgt& w) {
  float nrm = fsqrt_fast(__builtin_fmaf(id, id, iq * iq));

  // h = tanh(i_input @ W_psi_h + b_psi_h), W_psi_h is (3,4) row-major
  float h[4];
#pragma unroll
  for (int j = 0; j < 4; ++j) {
    float a = __builtin_fmaf(nrm, w.wph[8 + j],
              __builtin_fmaf(iq,  w.wph[4 + j],
              __builtin_fmaf(id,  w.wph[j], w.bph[j])));
    h[j] = ftanh(a);
  }

  // psi = h @ W_psi + b_psi, W_psi is (4,2) row-major
  float psd = w.bp[0], psq = w.bp[1];
#pragma unroll
  for (int j = 0; j < 4; ++j) {
    psd = __builtin_fmaf(h[j], w.wp[2 * j + 0], psd);
    psq = __builtin_fmaf(h[j], w.wp[2 * j + 1], psq);
  }

  // L = i_input @ W_L + b_L  ->  [[a,b],[c,d]]
  float L[4];
#pragma unroll
  for (int j = 0; j < 4; ++j) {
    L[j] = __builtin_fmaf(nrm, w.wl[8 + j],
           __builtin_fmaf(iq,  w.wl[4 + j],
           __builtin_fmaf(id,  w.wl[j], w.bl[j])));
  }
  float det  = __builtin_fmaf(L[0], L[3], -(L[1] * L[2]));
  float rdet = frcp_nr(det);

  // rhs @ vec with rhs = [[-Rs,-Rs, 0,  W*psi_q],
  //                       [-Rs,-Rs, S1,-W*psi_d]],  vec = [id, iq, u2, 1]
  float m  = -C_RS * (id + iq);
  float t0 = __builtin_fmaf(C_WEL, psq, m);
  float t1 = __builtin_fmaf(C_S1, u2, __builtin_fmaf(-C_WEL, psd, m));

  // inv_L @ t
  float k1d = __builtin_fmaf(L[3], t0, -(L[1] * t1)) * rdet;
  float k1q = __builtin_fmaf(L[0], t1, -(L[2] * t0)) * rdet;

  return make_float2(__builtin_fmaf(k1d, C_DT, id),
                     __builtin_fmaf(k1q, C_DT, iq));
}

// gfx1250 async global->LDS DMA of one 16B row per lane (ASYNCcnt-tracked).
__device__ __forceinline__ void async_row_load(uint32_t lds_addr, uint32_t byte_off,
                                               unsigned long long base) {
  asm volatile("global_load_async_to_lds_b128 %0, %1, %2"
               :: "v"(lds_addr), "v"(byte_off), "s"(base)
               : "memory");
}

__device__ __forceinline__ void wait_async_le(int n) {
  if (n <= 0)      asm volatile("s_wait_asynccnt 0x0" ::: "memory");
  else if (n == 1) asm volatile("s_wait_asynccnt 0x1" ::: "memory");
  else if (n == 2) asm volatile("s_wait_asynccnt 0x2" ::: "memory");
  else             asm volatile("s_wait_asynccnt 0x3" ::: "memory");
}

__global__ __launch_bounds__(TILE) void pmsm_stream_kernel(
    const float* __restrict__ inputs,
    const float* __restrict__ Wph, const float* __restrict__ bph,
    const float* __restrict__ Wp,  const float* __restrict__ bp,
    const float* __restrict__ WL,  const float* __restrict__ bL,
    float* __restrict__ out)
{
  __shared__ float4 buf[DEPTH][TILE];
  const int tid = threadIdx.x;

  Wgt w;
  load_weights(w, Wph, bph, Wp, bp, WL, bL);

  const unsigned long long base = (unsigned long long)(uintptr_t)inputs;
  const uint32_t row0 = (uint32_t)blockIdx.x * (uint32_t)(TPB * TILE) + (uint32_t)tid;

  // Per-lane private LDS slots: low 32 bits of the flat address = LDS offset.
  uint32_t lds_addr[DEPTH];
#pragma unroll
  for (int s = 0; s < DEPTH; ++s)
    lds_addr[s] = (uint32_t)(uintptr_t)&buf[s][tid];

  // Pipeline prologue: DEPTH-1 tiles in flight.
#pragma unroll
  for (int s = 0; s < DEPTH - 1; ++s)
    async_row_load(lds_addr[s], (row0 + (uint32_t)s * TILE) * 16u, base);

  float2* __restrict__ o = (float2*)out;
#pragma unroll
  for (int i = 0; i < TPB; ++i) {
    if (i + DEPTH - 1 < TPB)
      async_row_load(lds_addr[(i + DEPTH - 1) % DEPTH],
                     (row0 + (uint32_t)(i + DEPTH - 1) * TILE) * 16u, base);
    // loads complete in order: wait until tile i has landed in LDS
    wait_async_le(((i + DEPTH - 1) < TPB) ? (DEPTH - 1) : (TPB - 1 - i));

    float4 v = buf[i % DEPTH][tid];                     // ds_load_b128
    o[row0 + (uint32_t)i * TILE] =                      // global_store_b64
        compute_row(v.x, v.y, v.z, w);
  }
}

// Simple tail kernel for row counts not divisible by TILE*TPB (empty for B=4Mi).
__global__ __launch_bounds__(TILE) void pmsm_tail_kernel(
    const float* __restrict__ inputs,
    const float* __restrict__ Wph, const float* __restrict__ bph,
    const float* __restrict__ Wp,  const float* __restrict__ bp,
    const float* __restrict__ WL,  const float* __restrict__ bL,
    float* __restrict__ out, int startRow, int nRows)
{
  int row = startRow + (int)(blockIdx.x * blockDim.x + threadIdx.x);
  if (row >= nRows) return;
  Wgt w;
  load_weights(w, Wph, bph, Wp, bp, WL, bL);
  float4 v = ((const float4*)inputs)[row];
  ((float2*)out)[row] = compute_row(v.x, v.y, v.z, w);
}

extern "C" void kernel_launch(void* const* d_in, const int* in_sizes, int n_in,
                              void* d_out, int out_size, void* d_ws, size_t ws_size,
                              hipStream_t stream) {
  (void)n_in; (void)out_size; (void)d_ws; (void)ws_size;
  const float* inputs = (const float*)d_in[0];
  const float* Wph    = (const float*)d_in[1];
  const float* bph    = (const float*)d_in[2];
  const float* Wp     = (const float*)d_in[3];
  const float* bp     = (const float*)d_in[4];
  const float* WL     = (const float*)d_in[5];
  const float* bL     = (const float*)d_in[6];
  float* out = (float*)d_out;

  const long long rows         = (long long)in_sizes[0] / 4;
  const long long rowsPerBlock = (long long)TILE * TPB;
  const long long fullBlocks   = rows / rowsPerBlock;
  const long long startTail    = fullBlocks * rowsPerBlock;
  const long long tailRows     = rows - startTail;

  if (fullBlocks > 0) {
    pmsm_stream_kernel<<<(unsigned)fullBlocks, TILE, 0, stream>>>(
        inputs, Wph, bph, Wp, bp, WL, bL, out);
  }
  if (tailRows > 0) {
    unsigned tb = (unsigned)((tailRows + TILE - 1) / TILE);
    pmsm_tail_kernel<<<tb, TILE, 0, stream>>>(
        inputs, Wph, bph, Wp, bp, WL, bL, out, (int)startTail, (int)rows);
  }
}